// GraphConvBlock_50766513439198
// MI455X (gfx1250) — compile-verified
//
#include <hip/hip_runtime.h>

typedef float v2f __attribute__((ext_vector_type(2)));
typedef float v8f __attribute__((ext_vector_type(8)));

#define BATCH 4096
#define NV    64
#define DI    128
#define DO    128
#define BN_EPS 1e-5f

// LDS strides chosen for conflict-free WMMA fragment loads (dword units)
#define WSTRIDE 288   // packed W pair-rows: 2*DO + 32  (288 % 64 == 32)
#define FSTRIDE 132   // feat rows: DI + 4              (4*lane bank spread)
#define ASTRIDE 68    // adj rows: NV + 4
#define SSTRIDE 132   // s rows: DO + 4

// ---------------------------------------------------------------------------
// k0: zero the stats accumulators in workspace
// ---------------------------------------------------------------------------
__global__ void gcb_zero_kernel(float* __restrict__ p, int n) {
    int i = blockIdx.x * blockDim.x + threadIdx.x;
    if (i < n) p[i] = 0.0f;
}

// ---------------------------------------------------------------------------
// k1: per-vertex GEMM  h[b,v,:] = feat[b,v,:] @ W[v] + bias[v]
//     + accumulate per-(v,o) sum / sum-of-squares for BatchNorm.
// Grid: (32 batch-chunks of 128 rows, 64 vertices), 256 threads (8 waves).
// W is staged in LDS pre-packed into B-fragment pairs {W[k][n],W[k+1][n]}
// so each fragment is a single conflict-free ds_load_b64.
// ---------------------------------------------------------------------------
__global__ void gcb_gemm1_kernel(const float* __restrict__ feat,
                                 const float* __restrict__ W,
                                 const float* __restrict__ bias,
                                 float* __restrict__ h,
                                 float* __restrict__ gsum,
                                 float* __restrict__ gsq) {
    extern __shared__ float sm[];
    float* Ws   = sm;                         // 64 pair-rows * WSTRIDE
    float* Fs   = sm + 64 * WSTRIDE;          // 128 rows * FSTRIDE
    float* ssum = Fs + 128 * FSTRIDE;         // 128
    float* ssq  = ssum + 128;                 // 128

    const int v     = blockIdx.y;
    const int bbase = blockIdx.x * 128;
    const int tid   = threadIdx.x;

    if (tid < 128) { ssum[tid] = 0.0f; ssq[tid] = 0.0f; }

    // Stage W_v, packing rows (2k,2k+1) into interleaved pairs:
    // Ws[(i>>1)*WSTRIDE + o*2 + (i&1)] = W[i][o]
    {
        const float4* Wg = (const float4*)(W + (size_t)v * DI * DO);
        #pragma unroll 4
        for (int i = tid; i < DI * DO / 4; i += 256) {
            int r = i >> 5;              // W row 0..127
            int c = (i & 31) * 4;        // W col base
            float4 w = Wg[i];
            float* dst = &Ws[(r >> 1) * WSTRIDE + (r & 1)];
            dst[(c + 0) * 2] = w.x;
            dst[(c + 1) * 2] = w.y;
            dst[(c + 2) * 2] = w.z;
            dst[(c + 3) * 2] = w.w;
        }
    }
    // Stage 128 feat rows for this vertex (row-major, padded stride)
    {
        #pragma unroll 4
        for (int i = tid; i < 128 * DI / 4; i += 256) {
            int r = i >> 5;
            int c = (i & 31) * 4;
            *(float4*)(&Fs[r * FSTRIDE + c]) =
                *(const float4*)(feat + ((size_t)(bbase + r) * NV + v) * DI + c);
        }
    }
    __syncthreads();

    const int wave = tid >> 5;
    const int lane = tid & 31;
    const int m0   = wave * 16;
    const int nlo  = lane & 15;
    const int kk   = (lane >> 4) * 2;    // K-pair select per half-wave
    const int half = lane >> 4;

    v8f acc[8];
    #pragma unroll
    for (int nt = 0; nt < 8; ++nt) acc[nt] = (v8f)0.0f;

    const float* arow = &Fs[(m0 + nlo) * FSTRIDE];
    for (int k = 0; k < DI; k += 4) {
        v2f a = *(const v2f*)(arow + k + kk);                       // ds_load_b64
        const float* brow = &Ws[((k + kk) >> 1) * WSTRIDE + nlo * 2];
        #pragma unroll
        for (int nt = 0; nt < 8; ++nt) {
            v2f b = *(const v2f*)(brow + nt * 32);                  // ds_load_b64
            acc[nt] = __builtin_amdgcn_wmma_f32_16x16x4_f32(
                false, a, false, b, (short)0, acc[nt], false, false);
        }
    }

    // Epilogue: bias, store h, accumulate stats
    #pragma unroll
    for (int nt = 0; nt < 8; ++nt) {
        const int o  = nt * 16 + nlo;
        const float bv = bias[v * DO + o];
        float ps = 0.0f, pq = 0.0f;
        #pragma unroll
        for (int j = 0; j < 8; ++j) {
            float val = acc[nt][j] + bv;
            int row = bbase + m0 + j + half * 8;      // C layout: VGPR j -> M=j / j+8
            h[((size_t)row * NV + v) * DO + o] = val;
            ps += val; pq += val * val;
        }
        atomicAdd(&ssum[o], ps);   // ds_add_f32
        atomicAdd(&ssq[o], pq);
    }
    __syncthreads();
    if (tid < 128) {
        atomicAdd(&gsum[v * DO + tid], ssum[tid]);
        atomicAdd(&gsq[v * DO + tid], ssq[tid]);
    }
}

// ---------------------------------------------------------------------------
// k2a: per-channel BN affine fold: s = gamma*rsqrt(var+eps), t = beta - mu*s
// ---------------------------------------------------------------------------
__global__ void gcb_stats_kernel(const float* __restrict__ gsum,
                                 const float* __restrict__ gsq,
                                 const float* __restrict__ gamma,
                                 const float* __restrict__ beta,
                                 float* __restrict__ s,
                                 float* __restrict__ t) {
    int i = blockIdx.x * blockDim.x + threadIdx.x;
    if (i >= NV * DO) return;
    const float invB = 1.0f / (float)BATCH;
    float mu  = gsum[i] * invB;
    float var = gsq[i] * invB - mu * mu;
    float sc  = gamma[i] * rsqrtf(var + BN_EPS);
    s[i] = sc;
    t[i] = beta[i] - mu * sc;
}

// ---------------------------------------------------------------------------
// k2b: c[u,o] = sum_v adj[u,v] * t[v,o]
// ---------------------------------------------------------------------------
__global__ void gcb_cbias_kernel(const float* __restrict__ adj,
                                 const float* __restrict__ t,
                                 float* __restrict__ c) {
    int i = blockIdx.x * blockDim.x + threadIdx.x;
    if (i >= NV * DO) return;
    int u = i >> 7, o = i & 127;
    float acc = 0.0f;
    #pragma unroll 8
    for (int v = 0; v < NV; ++v) acc += adj[u * NV + v] * t[v * DO + o];
    c[i] = acc;
}

// ---------------------------------------------------------------------------
// k3: out[b,u,o] = relu( sum_v adj[u,v] * s[v,o] * h[b,v,o] + c[u,o] )
// GEMM per batch element: M=u(64), K=v(64), N=o(128) via wmma f32 16x16x4.
// 2048 WGs x 256 threads; wave = (batch, u-tile). adj + s staged in LDS
// with padded strides (conflict-free fragment loads).
// ---------------------------------------------------------------------------
__global__ void gcb_mix_kernel(const float* __restrict__ adj,
                               const float* __restrict__ h,
                               const float* __restrict__ s,
                               const float* __restrict__ c,
                               float* __restrict__ out) {
    __shared__ float adjs[NV * ASTRIDE];   // 17 KB
    __shared__ float ss[NV * SSTRIDE];     // 33 KB

    const int tid = threadIdx.x;
    #pragma unroll 2
    for (int i = tid; i < NV * NV / 4; i += 256) {
        int r = i >> 4, cc = (i & 15) * 4;
        *(float4*)(&adjs[r * ASTRIDE + cc]) = ((const float4*)adj)[i];
    }
    #pragma unroll 4
    for (int i = tid; i < NV * DO / 4; i += 256) {
        int r = i >> 5, cc = (i & 31) * 4;
        *(float4*)(&ss[r * SSTRIDE + cc]) = ((const float4*)s)[i];
    }
    __syncthreads();

    const int wave = tid >> 5;
    const int lane = tid & 31;
    const int b    = blockIdx.x * 2 + (wave >> 2);
    const int u0   = (wave & 3) * 16;
    const int nlo  = lane & 15;
    const int kk   = (lane >> 4) * 2;
    const int half = lane >> 4;

    v8f acc[8];
    #pragma unroll
    for (int nt = 0; nt < 8; ++nt) acc[nt] = (v8f)0.0f;

    const float* hb   = h + (size_t)b * NV * DO;
    const float* arow = &adjs[(u0 + nlo) * ASTRIDE];
    for (int k = 0; k < NV; k += 4) {
        v2f a = *(const v2f*)(arow + k + kk);          // ds_load_b64
        const float* h0 = hb + (k + kk) * DO + nlo;
        const float* s0 = &ss[(k + kk) * SSTRIDE + nlo];
        #pragma unroll
        for (int nt = 0; nt < 8; ++nt) {
            v2f bf;
            bf.x = h0[nt * 16]      * s0[nt * 16];
            bf.y = h0[nt * 16 + DO] * s0[nt * 16 + SSTRIDE];
            acc[nt] = __builtin_amdgcn_wmma_f32_16x16x4_f32(
                false, a, false, bf, (short)0, acc[nt], false, false);
        }
    }

    float* ob = out + (size_t)b * NV * DO;
    #pragma unroll
    for (int nt = 0; nt < 8; ++nt) {
        const int o = nt * 16 + nlo;
        #pragma unroll
        for (int j = 0; j < 8; ++j) {
            int u = u0 + j + half * 8;
            float val = acc[nt][j] + c[u * DO + o];
            ob[u * DO + o] = fmaxf(val, 0.0f);
        }
    }
}

// ---------------------------------------------------------------------------
// launch
// ---------------------------------------------------------------------------
extern "C" void kernel_launch(void* const* d_in, const int* in_sizes, int n_in,
                              void* d_out, int out_size, void* d_ws, size_t ws_size,
                              hipStream_t stream) {
    const float* feat  = (const float*)d_in[0];   // [4096,64,128]
    const float* adj   = (const float*)d_in[1];   // [64,64]
    const float* W     = (const float*)d_in[2];   // [64,128,128]
    const float* bias  = (const float*)d_in[3];   // [64,128]
    const float* gamma = (const float*)d_in[4];   // [64,128]
    const float* beta  = (const float*)d_in[5];   // [64,128]
    float* out = (float*)d_out;                   // [4096,64,128]

    float* ws   = (float*)d_ws;
    float* h    = ws;                              // 4096*64*128
    float* gsum = ws + (size_t)BATCH * NV * DO;    // 8192
    float* gsq  = gsum + NV * DO;                  // 8192
    float* s    = gsq + NV * DO;                   // 8192
    float* t    = s + NV * DO;                     // 8192
    float* c    = t + NV * DO;                     // 8192

    // zero stats
    gcb_zero_kernel<<<(2 * NV * DO + 255) / 256, 256, 0, stream>>>(gsum, 2 * NV * DO);

    // grouped GEMM + bias + stat accumulation
    dim3 g1(BATCH / 128, NV);
    size_t smem1 = (size_t)(64 * WSTRIDE + 128 * FSTRIDE + 256) * sizeof(float);
    gcb_gemm1_kernel<<<g1, 256, smem1, stream>>>(feat, W, bias, h, gsum, gsq);

    // BN fold + adjacency bias
    gcb_stats_kernel<<<(NV * DO + 255) / 256, 256, 0, stream>>>(gsum, gsq, gamma, beta, s, t);
    gcb_cbias_kernel<<<(NV * DO + 255) / 256, 256, 0, stream>>>(adj, t, c);

    // adjacency mix + relu
    gcb_mix_kernel<<<BATCH / 2, 256, 0, stream>>>(adj, h, s, c, out);
}